// GAT_2190433321454
// MI455X (gfx1250) — compile-verified
//
#include <hip/hip_runtime.h>
#include <hip/hip_bf16.h>
#include <math.h>

typedef __attribute__((ext_vector_type(2))) float v2f;
typedef __attribute__((ext_vector_type(8))) float v8f;

#define NNODES 100000
#define NEDGES 3200000
#define ETOT   (NEDGES + NNODES)   // reference appends one self-loop per node
#define FIN    128
#define HID    32
#define NCLS   40
#define NPAD   48                  // NCLS padded to 3 WMMA N-tiles
#define NEG_SLOPE 0.2f

// ---------- monotone float <-> uint mapping so atomicMax(u32) == float max ----
__device__ __forceinline__ unsigned f2ord(float f) {
    unsigned u = __float_as_uint(f);
    return (u & 0x80000000u) ? ~u : (u | 0x80000000u);
}
__device__ __forceinline__ float ord2f(unsigned u) {
    unsigned v = (u & 0x80000000u) ? (u & 0x7FFFFFFFu) : ~u;
    return __uint_as_float(v);
}

// ---------------- dense projection via V_WMMA_F32_16X16X4_F32 ----------------
// One wave -> 16 output rows x (NT*16) output cols.  X is [nrows,K] row-major,
// W is [K, NT*16] row-major (zero-padded cols beyond the real width),
// H is [nrows, NT*16].
// A 16x4 fragment: lane<16 row=lane holds K={k0,k0+1}; lane>=16 holds {k0+2,k0+3}.
// B 4x16 fragment mirrors A with N = lane&15.  C/D: VGPR r -> M = r + 8*(lane>=16).
template<int K, int NT>
__global__ __launch_bounds__(256)
void wmma_gemm_kernel(const float* __restrict__ X, const float* __restrict__ W,
                      float* __restrict__ H, int nrows) {
    constexpr int LD = NT * 16;
    const int wave = (int)((blockIdx.x * blockDim.x + threadIdx.x) >> 5);
    const int lane = (int)(threadIdx.x & 31);
    const int row_base = wave * 16;
    if (row_base >= nrows) return;            // wave-uniform: EXEC stays all-ones
    const int lh = lane >> 4;                 // 0 / 1 (K-half select)
    const int lm = lane & 15;                 // row / col within tile

    const v8f vzero = {0.f, 0.f, 0.f, 0.f, 0.f, 0.f, 0.f, 0.f};
    v8f acc[NT];
#pragma unroll
    for (int t = 0; t < NT; ++t) acc[t] = vzero;

    const float* xrow = X + (size_t)(row_base + lm) * K;
#pragma unroll 4
    for (int k0 = 0; k0 < K; k0 += 4) {
        const int ka = k0 + 2 * lh;
        v2f a; a[0] = xrow[ka]; a[1] = xrow[ka + 1];
#pragma unroll
        for (int t = 0; t < NT; ++t) {
            const float* wp = W + (size_t)ka * LD + t * 16 + lm;
            v2f b; b[0] = wp[0]; b[1] = wp[LD];
            acc[t] = __builtin_amdgcn_wmma_f32_16x16x4_f32(
                false, a, false, b, (short)0, acc[t], false, false);
        }
    }
#pragma unroll
    for (int t = 0; t < NT; ++t) {
#pragma unroll
        for (int r = 0; r < 8; ++r) {
            const int m = r + 8 * lh;
            H[(size_t)(row_base + m) * LD + t * 16 + lm] = acc[t][r];
        }
    }
}

// -------- per-node attention logits: as[i]=h[i].a_src, ad[i]=h[i].a_dst ------
template<int C, int LD>
__global__ void attn_scores_kernel(const float* __restrict__ h,
                                   const float* __restrict__ a_src,
                                   const float* __restrict__ a_dst,
                                   float* __restrict__ as_, float* __restrict__ ad_,
                                   int n) {
    int i = blockIdx.x * blockDim.x + threadIdx.x;
    if (i >= n) return;
    const float* hr = h + (size_t)i * LD;
    float s = 0.f, d = 0.f;
#pragma unroll
    for (int c = 0; c < C; ++c) { float v = hr[c]; s = fmaf(v, a_src[c], s); d = fmaf(v, a_dst[c], d); }
    as_[i] = s;
    ad_[i] = d;
}

// ---------------- edge pass 1: segment max of leaky_relu(as+ad) --------------
__global__ void edge_max_kernel(const int* __restrict__ src, const int* __restrict__ dst,
                                const float* __restrict__ as_, const float* __restrict__ ad_,
                                unsigned* __restrict__ menc) {
    int e = blockIdx.x * blockDim.x + threadIdx.x;
    if (e >= ETOT) return;
    int s, d;
    if (e < NEDGES) { s = src[e]; d = dst[e]; } else { s = d = e - NEDGES; }
    float v = as_[s] + ad_[d];
    v = (v > 0.f) ? v : NEG_SLOPE * v;
    atomicMax(menc + d, f2ord(v));
}

// ------- edge pass 2: ex = exp(e - m[dst]); denom[dst]+=ex; acc[dst]+=ex*h[src]
// One thread per (edge, 4-channel group): lane-adjacent threads write adjacent
// channels -> coalesced atomics within a cacheline, all L2-resident.
template<int C, int LD>
__global__ void edge_accum_kernel(const int* __restrict__ src, const int* __restrict__ dst,
                                  const float* __restrict__ as_, const float* __restrict__ ad_,
                                  const unsigned* __restrict__ menc,
                                  const float* __restrict__ h,
                                  float* __restrict__ denom, float* __restrict__ acc) {
    constexpr int G = C / 4;
    int tid = blockIdx.x * blockDim.x + threadIdx.x;
    int e = tid / G;
    int g = tid % G;
    if (e >= ETOT) return;
    int s, d;
    if (e < NEDGES) { s = src[e]; d = dst[e]; } else { s = d = e - NEDGES; }
    float v = as_[s] + ad_[d];
    v = (v > 0.f) ? v : NEG_SLOPE * v;
    float ex = expf(v - ord2f(menc[d]));
    if (g == 0) atomicAdd(denom + d, ex);
    const float* hr = h + (size_t)s * LD + g * 4;
    float* ar = acc + (size_t)d * C + g * 4;
#pragma unroll
    for (int j = 0; j < 4; ++j) atomicAdd(ar + j, ex * hr[j]);
}

// ---------------- layer-1 finalize: out1 = relu(acc/denom + b) ---------------
__global__ void finalize1_kernel(const float* __restrict__ acc, const float* __restrict__ denom,
                                 const float* __restrict__ b, float* __restrict__ out1, int n) {
    int i = blockIdx.x * blockDim.x + threadIdx.x;
    int node = i >> 5, c = i & 31;               // HID == 32
    if (node >= n) return;
    float v = acc[(size_t)node * HID + c] / denom[node] + b[c];
    out1[(size_t)node * HID + c] = (v > 0.f) ? v : 0.f;
}

// ---------------- zero-pad W2 [32x40] into [32x48] ---------------------------
__global__ void pad_w2_kernel(const float* __restrict__ W2, float* __restrict__ Wp) {
    int i = blockIdx.x * blockDim.x + threadIdx.x;
    if (i >= HID * NPAD) return;
    int r = i / NPAD, c = i % NPAD;
    Wp[i] = (c < NCLS) ? W2[r * NCLS + c] : 0.f;
}

// -------- layer-2 finalize: relu(acc/denom + b) then log_softmax over 40 -----
__global__ void finalize2_kernel(const float* __restrict__ acc, const float* __restrict__ denom,
                                 const float* __restrict__ b, float* __restrict__ out, int n) {
    int i = blockIdx.x * blockDim.x + threadIdx.x;
    if (i >= n) return;
    float t[NCLS];
    float inv = 1.f / denom[i];
    float mx = -INFINITY;
#pragma unroll
    for (int c = 0; c < NCLS; ++c) {
        float v = acc[(size_t)i * NCLS + c] * inv + b[c];
        v = (v > 0.f) ? v : 0.f;
        t[c] = v;
        mx = fmaxf(mx, v);
    }
    float s = 0.f;
#pragma unroll
    for (int c = 0; c < NCLS; ++c) s += expf(t[c] - mx);
    float ls = logf(s) + mx;
#pragma unroll
    for (int c = 0; c < NCLS; ++c) out[(size_t)i * NCLS + c] = t[c] - ls;
}

extern "C" void kernel_launch(void* const* d_in, const int* in_sizes, int n_in,
                              void* d_out, int out_size, void* d_ws, size_t ws_size,
                              hipStream_t stream) {
    const float* x    = (const float*)d_in[0];
    const int*   ei   = (const int*)  d_in[1];
    const float* W1   = (const float*)d_in[2];
    const float* asw1 = (const float*)d_in[3];
    const float* adw1 = (const float*)d_in[4];
    const float* b1   = (const float*)d_in[5];
    const float* W2   = (const float*)d_in[6];
    const float* asw2 = (const float*)d_in[7];
    const float* adw2 = (const float*)d_in[8];
    const float* b2   = (const float*)d_in[9];
    float* out = (float*)d_out;

    const int* src = ei;            // edge_index[0]
    const int* dst = ei + NEDGES;   // edge_index[1]

    const size_t N = NNODES;
    float* ws = (float*)d_ws;
    // ---- workspace bump allocation (floats) ----
    float*    h1   = ws;                           // 32N
    float*    as1  = ws + 32 * N;                  // N
    float*    ad1  = ws + 33 * N;                  // N
    unsigned* m1   = (unsigned*)(ws + 34 * N);     // N   }
    float*    den1 = ws + 35 * N;                  // N   } contiguous for memset
    float*    acc1 = ws + 36 * N;                  // 32N }
    float*    out1 = ws + 68 * N;                  // 32N
    float*    Wp2  = ws + 100 * N;                 // 2048 (uses 32*48)
    float*    h2   = ws + 100 * N + 2048;          // 48N
    float*    as2  = ws + 148 * N + 2048;          // N
    float*    ad2  = ws + 149 * N + 2048;          // N
    unsigned* m2   = (unsigned*)(ws + 150 * N + 2048); // N   }
    float*    den2 = ws + 151 * N + 2048;          // N   } contiguous for memset
    float*    acc2 = ws + 152 * N + 2048;          // 40N }

    const int TB = 256;
    // ================= layer 1 =================
    {   // h1 = x @ W1  (K=128, 2 N-tiles)
        int waves  = (NNODES + 15) / 16;
        int blocks = (waves * 32 + TB - 1) / TB;
        wmma_gemm_kernel<FIN, 2><<<blocks, TB, 0, stream>>>(x, W1, h1, NNODES);
    }
    attn_scores_kernel<HID, HID><<<(NNODES + TB - 1) / TB, TB, 0, stream>>>(
        h1, asw1, adw1, as1, ad1, NNODES);
    hipMemsetAsync(m1, 0, 34 * N * sizeof(float), stream);   // m1|den1|acc1 -> 0
    edge_max_kernel<<<(ETOT + TB - 1) / TB, TB, 0, stream>>>(src, dst, as1, ad1, m1);
    {
        long long tot = (long long)ETOT * (HID / 4);
        edge_accum_kernel<HID, HID><<<(int)((tot + TB - 1) / TB), TB, 0, stream>>>(
            src, dst, as1, ad1, m1, h1, den1, acc1);
    }
    finalize1_kernel<<<(NNODES * HID + TB - 1) / TB, TB, 0, stream>>>(
        acc1, den1, b1, out1, NNODES);

    // ================= layer 2 =================
    pad_w2_kernel<<<(HID * NPAD + TB - 1) / TB, TB, 0, stream>>>(W2, Wp2);
    {   // h2 = out1 @ W2pad  (K=32, 3 N-tiles, cols 40..47 are zero)
        int waves  = (NNODES + 15) / 16;
        int blocks = (waves * 32 + TB - 1) / TB;
        wmma_gemm_kernel<HID, 3><<<blocks, TB, 0, stream>>>(out1, Wp2, h2, NNODES);
    }
    attn_scores_kernel<NCLS, NPAD><<<(NNODES + TB - 1) / TB, TB, 0, stream>>>(
        h2, asw2, adw2, as2, ad2, NNODES);
    hipMemsetAsync(m2, 0, 42 * N * sizeof(float), stream);   // m2|den2|acc2 -> 0
    edge_max_kernel<<<(ETOT + TB - 1) / TB, TB, 0, stream>>>(src, dst, as2, ad2, m2);
    {
        long long tot = (long long)ETOT * (NCLS / 4);
        edge_accum_kernel<NCLS, NPAD><<<(int)((tot + TB - 1) / TB), TB, 0, stream>>>(
            src, dst, as2, ad2, m2, h2, den2, acc2);
    }
    finalize2_kernel<<<(NNODES + TB - 1) / TB, TB, 0, stream>>>(
        acc2, den2, b2, out, NNODES);
}